// ROBOLoss_14594298871844
// MI455X (gfx1250) — compile-verified
//
#include <hip/hip_runtime.h>
#include <math.h>

typedef float v2f __attribute__((ext_vector_type(2)));
typedef float v8f __attribute__((ext_vector_type(8)));

#define ROWS        4096
#define L_TOTAL     10000
#define GROUPS      2000            // L / 5
#define GROUP_TILES (GROUPS / 16)   // 125
#define ROW_CHUNKS  32              // row chunks per group tile
#define ROWS_PER_CHUNK (ROWS / ROW_CHUNKS)   // 128
#define WAVES_PER_BLOCK 8
#define NBLOCKS ((GROUP_TILES * ROW_CHUNKS) / WAVES_PER_BLOCK)  // 500
#define EPS_F   1e-5f
#define BBW     0.1f

__global__ void robo_zero_ws(float* __restrict__ ws, int n) {
    int i = blockIdx.x * blockDim.x + threadIdx.x;
    if (i < n) ws[i] = 0.0f;
}

// Phase 1: stream all (b,g) points; reduce num[g], den[g] over the batch axis
// with V_WMMA_F32_16X16X4_F32 (B = ones => D[m][n] = sum_k A[m][k] + C),
// and reduce the classification MSE with a wave shuffle.
__global__ __launch_bounds__(256) void robo_phase1(
        const float* __restrict__ inp,
        const float* __restrict__ tgt,
        float* __restrict__ ws_num,
        float* __restrict__ ws_den,
        float* __restrict__ ws_cls) {
    const int lane = threadIdx.x & 31;
    const int L    = lane & 15;       // group within tile / M row of A
    const int half = lane >> 4;       // K half: 0 -> K=0,1 ; 1 -> K=2,3
    const int wid  = blockIdx.x * WAVES_PER_BLOCK + (threadIdx.x >> 5);
    const int gt    = wid / ROW_CHUNKS;       // group tile [0,125)
    const int chunk = wid % ROW_CHUNKS;       // row chunk  [0,32)

    const int    g    = gt * 16 + L;
    const size_t gofs = (size_t)g * 5;

    v2f ones; ones.x = 1.0f; ones.y = 1.0f;  // B matrix: all ones (layout-free)
    v8f acc_num = {};
    v8f acc_den = {};
    float cls = 0.0f;

    const int b0 = chunk * ROWS_PER_CHUNK;
    #pragma unroll 4
    for (int bb = 0; bb < ROWS_PER_CHUNK; bb += 4) {
        const int rb = b0 + bb + 2 * half;   // this lane's first batch row
        v2f a_num, a_den;
        {   // row rb  -> A element K in VGPR0
            const float* ip = inp + (size_t)rb * L_TOTAL + gofs;
            const float* tp = tgt + (size_t)rb * L_TOTAL + gofs;
            float s  = ip[0];
            float ct = (tp[0] > 0.0f) ? 1.0f : 0.0f;
            float d1 = ip[1] - tp[1], d2 = ip[2] - tp[2];
            float d3 = ip[3] - tp[3], d4 = ip[4] - tp[4];
            float bberr = 0.25f * (d1*d1 + d2*d2 + d3*d3 + d4*d4);
            // sigmoid via single v_exp_f32 + v_rcp_f32 (avoid IEEE div expansion)
            float e = __builtin_amdgcn_rcpf(1.0f + __expf(-s)) - ct;
            cls = fmaf(e, e, cls);
            a_num.x = ct * bberr;
            a_den.x = ct;
        }
        {   // row rb+1 -> A element K in VGPR1
            const float* ip = inp + (size_t)(rb + 1) * L_TOTAL + gofs;
            const float* tp = tgt + (size_t)(rb + 1) * L_TOTAL + gofs;
            float s  = ip[0];
            float ct = (tp[0] > 0.0f) ? 1.0f : 0.0f;
            float d1 = ip[1] - tp[1], d2 = ip[2] - tp[2];
            float d3 = ip[3] - tp[3], d4 = ip[4] - tp[4];
            float bberr = 0.25f * (d1*d1 + d2*d2 + d3*d3 + d4*d4);
            float e = __builtin_amdgcn_rcpf(1.0f + __expf(-s)) - ct;
            cls = fmaf(e, e, cls);
            a_num.y = ct * bberr;
            a_den.y = ct;
        }
        // D[m][*] += sum_k A[m][k] : batch-axis reduction on the matrix pipe
        acc_num = __builtin_amdgcn_wmma_f32_16x16x4_f32(
            false, a_num, false, ones, (short)0, acc_num, false, false);
        acc_den = __builtin_amdgcn_wmma_f32_16x16x4_f32(
            false, a_den, false, ones, (short)0, acc_den, false, false);
    }

    // classification partial: wave32 reduce, one atomic per wave
    #pragma unroll
    for (int off = 16; off > 0; off >>= 1)
        cls += __shfl_down(cls, off, 32);
    if (lane == 0) atomicAdd(ws_cls, cls);

    // C/D layout: VGPR r, lane 0 -> D[r][0]; VGPR r, lane 16 -> D[r+8][0]
    if (L == 0) {
        #pragma unroll
        for (int r = 0; r < 8; ++r) {
            const int gg = gt * 16 + half * 8 + r;
            atomicAdd(&ws_num[gg], acc_num[r]);
            atomicAdd(&ws_den[gg], acc_den[r]);
        }
    }
}

// Phase 2: single block combines the 2000-entry num/den arrays + cls scalar.
__global__ __launch_bounds__(256) void robo_finalize(
        const float* __restrict__ ws_num,
        const float* __restrict__ ws_den,
        const float* __restrict__ ws_cls,
        float* __restrict__ out) {
    __shared__ float red[256];
    float s = 0.0f;
    for (int g = threadIdx.x; g < GROUPS; g += 256)
        s += ws_num[g] / (ws_den[g] + EPS_F);
    red[threadIdx.x] = s;
    __syncthreads();
    for (int st = 128; st > 0; st >>= 1) {
        if (threadIdx.x < st) red[threadIdx.x] += red[threadIdx.x + st];
        __syncthreads();
    }
    if (threadIdx.x == 0) {
        float class_loss = ws_cls[0] / (float)(ROWS * GROUPS);
        out[0] = class_loss + BBW * red[0];
    }
}

extern "C" void kernel_launch(void* const* d_in, const int* in_sizes, int n_in,
                              void* d_out, int out_size, void* d_ws, size_t ws_size,
                              hipStream_t stream) {
    const float* inp = (const float*)d_in[0];
    const float* tgt = (const float*)d_in[1];
    float* ws     = (float*)d_ws;
    float* ws_num = ws;                 // [0, 2000)
    float* ws_den = ws + GROUPS;        // [2000, 4000)
    float* ws_cls = ws + 2 * GROUPS;    // [4000]

    robo_zero_ws<<<16, 256, 0, stream>>>(ws, 2 * GROUPS + 1);
    robo_phase1<<<NBLOCKS, 256, 0, stream>>>(inp, tgt, ws_num, ws_den, ws_cls);
    robo_finalize<<<1, 256, 0, stream>>>(ws_num, ws_den, ws_cls, (float*)d_out);
}